// CQAttention_18820546691785
// MI455X (gfx1250) — compile-verified
//
#include <hip/hip_runtime.h>
#include <cmath>

#define BATCH 64
#define LCDIM 1024
#define LQDIM 128
#define DDIM  128
#define NEGV  (-1e9f)

typedef __attribute__((ext_vector_type(2))) float v2f;
typedef __attribute__((ext_vector_type(8))) float v8f;

__device__ __forceinline__ v8f wmma_f32_k4(v2f a, v2f b, v8f c) {
  // D = A(16x4,f32) * B(4x16,f32) + C(16x16,f32)   -> v_wmma_f32_16x16x4_f32
  return __builtin_amdgcn_wmma_f32_16x16x4_f32(false, a, false, b, (short)0, c,
                                               false, false);
}

// ---------------------------------------------------------------------------
// Kernel 1: raw scores S = (C*w_mul)@Q^T + cc + cq + bias ; also out[:,0:D]=C
// grid (LC/64, B), block 256 (8 waves)
// ---------------------------------------------------------------------------
__global__ __launch_bounds__(256) void k_scores(
    const float* __restrict__ Cg, const float* __restrict__ Qg,
    const float* __restrict__ wc, const float* __restrict__ wq,
    const float* __restrict__ wm, const float* __restrict__ bias,
    float* __restrict__ Sraw, float* __restrict__ out) {
  __shared__ float Qs[LQDIM][DDIM + 4];   // Q rows (B operand = Q^T: read Qs[n][k])
  __shared__ float Chs[64][DDIM + 4];     // Chat = C*w_mul tile
  __shared__ float wcl[DDIM], wql[DDIM], wml[DDIM];
  __shared__ float ccs[64], cqs[LQDIM];
  __shared__ float red[256];

  const int t = threadIdx.x;
  const int b = blockIdx.y;
  const int i0 = blockIdx.x * 64;

  if (t < DDIM) { wcl[t] = wc[t]; wql[t] = wq[t]; wml[t] = wm[t]; }
  __syncthreads();

  // load C tile (64 x 128): emit out[...,0:D]=C, build Chat, partial cc
  {
    const int r = t >> 2, seg = t & 3;
    const float* src = Cg + ((size_t)b * LCDIM + i0 + r) * DDIM + seg * 32;
    float* dst = out + ((size_t)b * LCDIM + i0 + r) * (4 * DDIM) + seg * 32;
    float cc = 0.f;
#pragma unroll
    for (int u = 0; u < 8; ++u) {
      float4 v = reinterpret_cast<const float4*>(src)[u];
      reinterpret_cast<float4*>(dst)[u] = v;
      const int d = seg * 32 + u * 4;
      Chs[r][d + 0] = v.x * wml[d + 0];
      Chs[r][d + 1] = v.y * wml[d + 1];
      Chs[r][d + 2] = v.z * wml[d + 2];
      Chs[r][d + 3] = v.w * wml[d + 3];
      cc += v.x * wcl[d + 0] + v.y * wcl[d + 1] + v.z * wcl[d + 2] +
            v.w * wcl[d + 3];
    }
    red[t] = cc;
  }
  __syncthreads();
  if (t < 64) ccs[t] = red[4 * t] + red[4 * t + 1] + red[4 * t + 2] + red[4 * t + 3];

  // load Q (128 x 128), partial cq
  float cqpart = 0.f;
  {
    const int j = t >> 1, h = t & 1;
    const float* src = Qg + ((size_t)b * LQDIM + j) * DDIM + h * 64;
#pragma unroll
    for (int u = 0; u < 16; ++u) {
      float4 v = reinterpret_cast<const float4*>(src)[u];
      const int d = h * 64 + u * 4;
      Qs[j][d + 0] = v.x; Qs[j][d + 1] = v.y;
      Qs[j][d + 2] = v.z; Qs[j][d + 3] = v.w;
      cqpart += v.x * wql[d + 0] + v.y * wql[d + 1] + v.z * wql[d + 2] +
                v.w * wql[d + 3];
    }
  }
  __syncthreads();
  red[t] = cqpart;
  __syncthreads();
  if (t < LQDIM) cqs[t] = red[2 * t] + red[2 * t + 1];
  __syncthreads();

  // WMMA: S tile (64 x 128). wave w: row-tile mt=w>>1, col-tiles (w&1)*4 .. +3
  const float bv = bias[0];
  const int lane = t & 31, w = t >> 5;
  const int mt = w >> 1;
  const int mloc = lane & 15;
  const int koff = (lane >> 4) * 2;
  const int hi = (lane >> 4) * 8;
  const int m = mt * 16 + mloc;

  v2f areg[32];                       // A fragments reused across 4 col-tiles
#pragma unroll
  for (int k0 = 0; k0 < DDIM; k0 += 4) {
    areg[k0 >> 2].x = Chs[m][k0 + koff];
    areg[k0 >> 2].y = Chs[m][k0 + koff + 1];
  }
#pragma unroll
  for (int q = 0; q < 4; ++q) {
    const int nt = (w & 1) * 4 + q;
    const int n = nt * 16 + mloc;
    v8f acc = {0.f, 0.f, 0.f, 0.f, 0.f, 0.f, 0.f, 0.f};
#pragma unroll
    for (int k0 = 0; k0 < DDIM; k0 += 4) {
      v2f bb;
      bb.x = Qs[n][k0 + koff];
      bb.y = Qs[n][k0 + koff + 1];
      acc = wmma_f32_k4(areg[k0 >> 2], bb, acc);
    }
#pragma unroll
    for (int v = 0; v < 8; ++v) {
      const int r = mt * 16 + v + hi;
      Sraw[((size_t)b * LCDIM + i0 + r) * LQDIM + n] =
          acc[v] + ccs[r] + cqs[n] + bv;
    }
  }
}

// ---------------------------------------------------------------------------
// Kernel 2a: partial column softmax stats (online max/sum over 128-row slices)
// grid (8, B), block 128 (thread = column j)
// ---------------------------------------------------------------------------
__global__ __launch_bounds__(128) void k_colstat_part(
    const float* __restrict__ Sraw, const int* __restrict__ Cmask,
    float* __restrict__ pm, float* __restrict__ ps) {
  const int j = threadIdx.x, s = blockIdx.x, b = blockIdx.y;
  float m = -INFINITY, sum = 0.f;
  for (int ii = 0; ii < 128; ++ii) {
    const int i = s * 128 + ii;
    float x;
    if (Cmask[b * LCDIM + i] != 0)
      x = Sraw[((size_t)b * LCDIM + i) * LQDIM + j];
    else
      x = NEGV;
    const float mn = fmaxf(m, x);
    sum = sum * __expf(m - mn) + __expf(x - mn);
    m = mn;
  }
  pm[((size_t)b * 8 + s) * LQDIM + j] = m;
  ps[((size_t)b * 8 + s) * LQDIM + j] = sum;
}

// Kernel 2b: combine 8 partials per column. grid B, block 128
__global__ __launch_bounds__(128) void k_colstat_comb(
    const float* __restrict__ pm, const float* __restrict__ ps,
    float* __restrict__ cmax, float* __restrict__ csum) {
  const int j = threadIdx.x, b = blockIdx.x;
  float m = -INFINITY;
#pragma unroll
  for (int s = 0; s < 8; ++s) m = fmaxf(m, pm[((size_t)b * 8 + s) * LQDIM + j]);
  float sum = 0.f;
#pragma unroll
  for (int s = 0; s < 8; ++s)
    sum += ps[((size_t)b * 8 + s) * LQDIM + j] *
           __expf(pm[((size_t)b * 8 + s) * LQDIM + j] - m);
  cmax[b * LQDIM + j] = m;
  csum[b * LQDIM + j] = sum;
}

// ---------------------------------------------------------------------------
// Kernel 3: T = S2^T @ C  per batch ([128 x 128], K over i=1024)
// grid B, block 256 (wave w owns j-tile w; 8 d-tiles each)
// ---------------------------------------------------------------------------
__global__ __launch_bounds__(256) void k_T(
    const float* __restrict__ Sraw, const float* __restrict__ Cg,
    const int* __restrict__ Cmask, const float* __restrict__ cmax,
    const float* __restrict__ csum, float* __restrict__ Tg) {
  __shared__ float S2T[LQDIM][65];   // [j][i_local], pad 65 (odd) for banks
  __shared__ float Cs[64][DDIM + 4]; // [i_local][d]
  __shared__ float cm[LQDIM], cs[LQDIM];

  const int t = threadIdx.x, b = blockIdx.x;
  const int lane = t & 31, w = t >> 5;
  const int mloc = lane & 15;
  const int koff = (lane >> 4) * 2;
  const int jrow = w * 16 + mloc;

  if (t < LQDIM) { cm[t] = cmax[b * LQDIM + t]; cs[t] = csum[b * LQDIM + t]; }

  v8f acc[8];
#pragma unroll
  for (int dt = 0; dt < 8; ++dt)
    acc[dt] = (v8f){0.f, 0.f, 0.f, 0.f, 0.f, 0.f, 0.f, 0.f};

  for (int ch = 0; ch < 16; ++ch) {
    __syncthreads();  // previous compute done; cm/cs ready on first pass
    const int r = t >> 2, seg = t & 3;
    const int i = ch * 64 + r;
    const int msk = Cmask[b * LCDIM + i];
    const float* srow = Sraw + ((size_t)b * LCDIM + i) * LQDIM + seg * 32;
    const float* crow = Cg + ((size_t)b * LCDIM + i) * DDIM + seg * 32;
#pragma unroll
    for (int u = 0; u < 8; ++u) {
      float4 sv = reinterpret_cast<const float4*>(srow)[u];
      float4 cv = reinterpret_cast<const float4*>(crow)[u];
      const int j = seg * 32 + u * 4;
      const float x0 = msk ? sv.x : NEGV;
      const float x1 = msk ? sv.y : NEGV;
      const float x2 = msk ? sv.z : NEGV;
      const float x3 = msk ? sv.w : NEGV;
      S2T[j + 0][r] = __expf(x0 - cm[j + 0]) / cs[j + 0];
      S2T[j + 1][r] = __expf(x1 - cm[j + 1]) / cs[j + 1];
      S2T[j + 2][r] = __expf(x2 - cm[j + 2]) / cs[j + 2];
      S2T[j + 3][r] = __expf(x3 - cm[j + 3]) / cs[j + 3];
      Cs[r][j + 0] = cv.x; Cs[r][j + 1] = cv.y;
      Cs[r][j + 2] = cv.z; Cs[r][j + 3] = cv.w;
    }
    __syncthreads();

    v2f areg[16];  // A fragments (this wave's j-tile), reused for 8 d-tiles
#pragma unroll
    for (int k0 = 0; k0 < 64; k0 += 4) {
      areg[k0 >> 2].x = S2T[jrow][k0 + koff];
      areg[k0 >> 2].y = S2T[jrow][k0 + koff + 1];
    }
#pragma unroll
    for (int dt = 0; dt < 8; ++dt) {
      const int n = dt * 16 + mloc;
#pragma unroll
      for (int k0 = 0; k0 < 64; k0 += 4) {
        v2f bb;
        bb.x = Cs[k0 + koff][n];
        bb.y = Cs[k0 + koff + 1][n];
        acc[dt] = wmma_f32_k4(areg[k0 >> 2], bb, acc[dt]);
      }
    }
  }

  const int hi = (lane >> 4) * 8;
#pragma unroll
  for (int dt = 0; dt < 8; ++dt) {
#pragma unroll
    for (int v = 0; v < 8; ++v) {
      const int j = w * 16 + v + hi;
      const int d = dt * 16 + mloc;
      Tg[((size_t)b * LQDIM + j) * DDIM + d] = acc[dt][v];
    }
  }
}

// ---------------------------------------------------------------------------
// Kernel 4: S1 = row-softmax(S, Qmask); A = S1@Q ; Bm = S1@T ;
//           out[:,D:2D]=A, out[:,2D:3D]=C*A, out[:,3D:4D]=C*Bm
// grid (LC/64, B), block 256
// ---------------------------------------------------------------------------
__global__ __launch_bounds__(256) void k_final(
    const float* __restrict__ Sraw, const float* __restrict__ Cg,
    const float* __restrict__ Qg, const float* __restrict__ Tg,
    const int* __restrict__ Qmask, float* __restrict__ out) {
  __shared__ float S1s[64][LQDIM + 4];
  __shared__ float Bs[LQDIM][DDIM + 4];  // holds Q, then T
  __shared__ float qmf[LQDIM];
  __shared__ float rowm[64], rowsum[64];
  __shared__ float redm[256], reds[256];

  const int t = threadIdx.x;
  const int b = blockIdx.y;
  const int i0 = blockIdx.x * 64;
  const int r = t >> 2, seg = t & 3;

  if (t < LQDIM) qmf[t] = (Qmask[b * LQDIM + t] != 0) ? 1.f : 0.f;
  __syncthreads();

  // load S tile, apply Qmask, online per-thread softmax stats over 32 cols
  {
    const float* srow = Sraw + ((size_t)b * LCDIM + i0 + r) * LQDIM + seg * 32;
    float m = -INFINITY, s = 0.f;
#pragma unroll
    for (int u = 0; u < 8; ++u) {
      float4 v = reinterpret_cast<const float4*>(srow)[u];
      const int j = seg * 32 + u * 4;
      const float x0 = (qmf[j + 0] != 0.f) ? v.x : NEGV;
      const float x1 = (qmf[j + 1] != 0.f) ? v.y : NEGV;
      const float x2 = (qmf[j + 2] != 0.f) ? v.z : NEGV;
      const float x3 = (qmf[j + 3] != 0.f) ? v.w : NEGV;
      S1s[r][j + 0] = x0; S1s[r][j + 1] = x1;
      S1s[r][j + 2] = x2; S1s[r][j + 3] = x3;
      const float mx = fmaxf(fmaxf(x0, x1), fmaxf(x2, x3));
      const float mn = fmaxf(m, mx);
      s = s * __expf(m - mn) + __expf(x0 - mn) + __expf(x1 - mn) +
          __expf(x2 - mn) + __expf(x3 - mn);
      m = mn;
    }
    redm[t] = m; reds[t] = s;
  }
  __syncthreads();
  if (t < 64) {
    float m = redm[4 * t];
    m = fmaxf(m, redm[4 * t + 1]);
    m = fmaxf(m, redm[4 * t + 2]);
    m = fmaxf(m, redm[4 * t + 3]);
    float s = 0.f;
#pragma unroll
    for (int k = 0; k < 4; ++k) s += reds[4 * t + k] * __expf(redm[4 * t + k] - m);
    rowm[t] = m; rowsum[t] = s;
  }
  __syncthreads();
  {
    const float m = rowm[r], inv = 1.f / rowsum[r];
#pragma unroll
    for (int u = 0; u < 8; ++u) {
      const int j = seg * 32 + u * 4;
      S1s[r][j + 0] = __expf(S1s[r][j + 0] - m) * inv;
      S1s[r][j + 1] = __expf(S1s[r][j + 1] - m) * inv;
      S1s[r][j + 2] = __expf(S1s[r][j + 2] - m) * inv;
      S1s[r][j + 3] = __expf(S1s[r][j + 3] - m) * inv;
    }
  }
  // load Q into Bs
  {
    const int j = t >> 1, h = t & 1;
    const float* src = Qg + ((size_t)b * LQDIM + j) * DDIM + h * 64;
#pragma unroll
    for (int u = 0; u < 16; ++u) {
      float4 v = reinterpret_cast<const float4*>(src)[u];
      const int d = h * 64 + u * 4;
      Bs[j][d + 0] = v.x; Bs[j][d + 1] = v.y;
      Bs[j][d + 2] = v.z; Bs[j][d + 3] = v.w;
    }
  }
  __syncthreads();

  const int lane = t & 31, w = t >> 5;
  const int mt = w >> 1;
  const int mloc = lane & 15;
  const int koff = (lane >> 4) * 2;
  const int hi = (lane >> 4) * 8;
  const int mrow = mt * 16 + mloc;

  v2f areg[32];  // S1 A-fragments: reused for BOTH matmuls
#pragma unroll
  for (int k0 = 0; k0 < LQDIM; k0 += 4) {
    areg[k0 >> 2].x = S1s[mrow][k0 + koff];
    areg[k0 >> 2].y = S1s[mrow][k0 + koff + 1];
  }

  // A = S1 @ Q
#pragma unroll
  for (int q = 0; q < 4; ++q) {
    const int nt = (w & 1) * 4 + q;
    const int n = nt * 16 + mloc;
    v8f acc = {0.f, 0.f, 0.f, 0.f, 0.f, 0.f, 0.f, 0.f};
#pragma unroll
    for (int k0 = 0; k0 < LQDIM; k0 += 4) {
      v2f bb;
      bb.x = Bs[k0 + koff][n];
      bb.y = Bs[k0 + koff + 1][n];
      acc = wmma_f32_k4(areg[k0 >> 2], bb, acc);
    }
#pragma unroll
    for (int v = 0; v < 8; ++v) {
      const int rr = mt * 16 + v + hi;
      const float cval = Cg[((size_t)b * LCDIM + i0 + rr) * DDIM + n];
      float* orow = out + ((size_t)b * LCDIM + i0 + rr) * (4 * DDIM);
      orow[DDIM + n] = acc[v];
      orow[2 * DDIM + n] = cval * acc[v];
    }
  }
  __syncthreads();
  // load T into Bs
  {
    const int j = t >> 1, h = t & 1;
    const float* src = Tg + ((size_t)b * LQDIM + j) * DDIM + h * 64;
#pragma unroll
    for (int u = 0; u < 16; ++u) {
      float4 v = reinterpret_cast<const float4*>(src)[u];
      const int d = h * 64 + u * 4;
      Bs[j][d + 0] = v.x; Bs[j][d + 1] = v.y;
      Bs[j][d + 2] = v.z; Bs[j][d + 3] = v.w;
    }
  }
  __syncthreads();
  // Bm = S1 @ T
#pragma unroll
  for (int q = 0; q < 4; ++q) {
    const int nt = (w & 1) * 4 + q;
    const int n = nt * 16 + mloc;
    v8f acc = {0.f, 0.f, 0.f, 0.f, 0.f, 0.f, 0.f, 0.f};
#pragma unroll
    for (int k0 = 0; k0 < LQDIM; k0 += 4) {
      v2f bb;
      bb.x = Bs[k0 + koff][n];
      bb.y = Bs[k0 + koff + 1][n];
      acc = wmma_f32_k4(areg[k0 >> 2], bb, acc);
    }
#pragma unroll
    for (int v = 0; v < 8; ++v) {
      const int rr = mt * 16 + v + hi;
      const float cval = Cg[((size_t)b * LCDIM + i0 + rr) * DDIM + n];
      float* orow = out + ((size_t)b * LCDIM + i0 + rr) * (4 * DDIM);
      orow[3 * DDIM + n] = cval * acc[v];
    }
  }
}

// ---------------------------------------------------------------------------
extern "C" void kernel_launch(void* const* d_in, const int* in_sizes, int n_in,
                              void* d_out, int out_size, void* d_ws,
                              size_t ws_size, hipStream_t stream) {
  (void)in_sizes; (void)n_in; (void)out_size; (void)ws_size;
  const float* Cg   = (const float*)d_in[0];
  const float* Qg   = (const float*)d_in[1];
  const int*   Cmk  = (const int*)d_in[2];
  const int*   Qmk  = (const int*)d_in[3];
  const float* wc   = (const float*)d_in[4];
  const float* wq   = (const float*)d_in[5];
  const float* wm   = (const float*)d_in[6];
  const float* bias = (const float*)d_in[7];
  float* out = (float*)d_out;

  // workspace layout (floats): ~38.3 MB total, fits comfortably in 192MB L2
  float* Sraw = (float*)d_ws;                                  // B*LC*LQ
  float* pm   = Sraw + (size_t)BATCH * LCDIM * LQDIM;          // B*8*LQ
  float* ps   = pm + (size_t)BATCH * 8 * LQDIM;                // B*8*LQ
  float* cmax = ps + (size_t)BATCH * 8 * LQDIM;                // B*LQ
  float* csum = cmax + (size_t)BATCH * LQDIM;                  // B*LQ
  float* Tg   = csum + (size_t)BATCH * LQDIM;                  // B*LQ*D

  k_scores<<<dim3(LCDIM / 64, BATCH), 256, 0, stream>>>(Cg, Qg, wc, wq, wm,
                                                        bias, Sraw, out);
  k_colstat_part<<<dim3(8, BATCH), 128, 0, stream>>>(Sraw, Cmk, pm, ps);
  k_colstat_comb<<<BATCH, 128, 0, stream>>>(pm, ps, cmax, csum);
  k_T<<<BATCH, 256, 0, stream>>>(Sraw, Cg, Cmk, cmax, csum, Tg);
  k_final<<<dim3(LCDIM / 64, BATCH), 256, 0, stream>>>(Sraw, Cg, Qg, Tg, Qmk,
                                                       out);
}